// EarlyFusion_11819749999186
// MI455X (gfx1250) — compile-verified
//
#include <hip/hip_runtime.h>
#include <cstdint>
#include <cstddef>

// ---------------------------------------------------------------------------
// EarlyFusion for gfx1250 (MI455X).
// Pure streaming op (~220 MB traffic, ~0.3 GFLOP) -> bandwidth bound.
// CDNA5 path used: GLOBAL_LOAD_ASYNC_TO_LDS_B128 (ASYNCcnt DMA) to stage the
// two source image rows per channel in LDS, s_wait_asynccnt, ds_load_b128
// reads, and nontemporal global_store_b128 for the 168 MB output stream.
// ---------------------------------------------------------------------------

typedef float v4f __attribute__((ext_vector_type(4)));

#define ENC_S   512
#define IN_DIM  1024
#define NQ      8
#define NP      4

__device__ __forceinline__ void async_load_b128_to_lds(unsigned lds_addr, const void* gaddr) {
    // gfx1250: copy 16 bytes per lane, global memory -> LDS, tracked by ASYNCcnt.
    asm volatile("global_load_async_to_lds_b128 %0, %1, off"
                 :: "v"(lds_addr), "v"((unsigned long long)(uintptr_t)gaddr)
                 : "memory");
}

__device__ __forceinline__ void wait_asynccnt0() {
#if __has_builtin(__builtin_amdgcn_s_wait_asynccnt)
    __builtin_amdgcn_s_wait_asynccnt(0);
#else
    asm volatile("s_wait_asynccnt 0" ::: "memory");
#endif
}

__global__ __launch_bounds__(128)
void early_fusion_kernel(const float* __restrict__ img,     // [B,3,1024,1024]
                         const float* __restrict__ pts,     // [B,8,4,2]
                         const int*   __restrict__ labels,  // [B,8,4]
                         const float* __restrict__ mean,    // [3]
                         const float* __restrict__ stdv,    // [3]
                         float* __restrict__ out)           // [B*8,5,512,512]
{
    __shared__ __align__(16) float smem[3][2][IN_DIM];   // 24 KB: 3 ch x 2 src rows

    const int tid  = (int)threadIdx.x;     // 0..127
    const int lane = tid & 31;             // wave32 lane
    const int wv   = tid >> 5;             // wave index 0..3
    const int y    = (int)blockIdx.x;      // output row 0..511
    const int b    = (int)blockIdx.y;      // batch

    // ---- 1) Kick off async DMA of source rows 2y, 2y+1 (all 3 channels). ----
    // Wave wv owns source float columns [256*wv, 256*wv+256): it loads exactly
    // the LDS slice it will read, so s_wait_asynccnt alone synchronizes it.
    const unsigned lbase = (unsigned)(uintptr_t)(&smem[0][0][0]);
    #pragma unroll
    for (int c = 0; c < 3; ++c) {
        #pragma unroll
        for (int r = 0; r < 2; ++r) {
            const float* grow = img + (((size_t)b * 3 + c) * IN_DIM + (size_t)(2 * y + r)) * IN_DIM;
            const unsigned lrow = lbase + (unsigned)(((c * 2 + r) * IN_DIM) * sizeof(float));
            #pragma unroll
            for (int k = 0; k < 2; ++k) {
                const unsigned fofs = (unsigned)(256 * wv + 128 * k + 4 * lane); // float index
                async_load_b128_to_lds(lrow + fofs * 4u, grow + fofs);
            }
        }
    }

    // ---- 2) Overlap: normalization constants (tiny, L2-resident). ----
    float mean_c[3], inv_std[3];
    #pragma unroll
    for (int c = 0; c < 3; ++c) {
        mean_c[c]  = mean[c];
        inv_std[c] = 1.0f / stdv[c];
    }

    // ---- 3) Wait for DMA, compute normalized 2x2-average image pixels. ----
    wait_asynccnt0();

    const int x0 = tid * 4;                 // 4 output pixels per thread
    v4f ch[3];
    #pragma unroll
    for (int c = 0; c < 3; ++c) {
        const v4f r0a = *(const v4f*)&smem[c][0][2 * x0];
        const v4f r0b = *(const v4f*)&smem[c][0][2 * x0 + 4];
        const v4f r1a = *(const v4f*)&smem[c][1][2 * x0];
        const v4f r1b = *(const v4f*)&smem[c][1][2 * x0 + 4];
        v4f v;
        v.x = r0a.x + r0a.y + r1a.x + r1a.y;
        v.y = r0a.z + r0a.w + r1a.z + r1a.w;
        v.z = r0b.x + r0b.y + r1b.x + r1b.y;
        v.w = r0b.z + r0b.w + r1b.z + r1b.w;
        ch[c] = (v * 0.25f - mean_c[c]) * inv_std[c];
    }

    // ---- 4) Per-query masks + broadcast stores (NT: 168 MB streamed out). ----
    const float yf  = (float)y;
    const float xf0 = (float)x0;

    #pragma unroll 1
    for (int q = 0; q < NQ; ++q) {
        const int bq = b * NQ + q;
        const float* pp = pts    + (size_t)bq * NP * 2;
        const int*   ll = labels + (size_t)bq * NP;

        float px[NP], py[NP];
        int   lab[NP];
        #pragma unroll
        for (int p = 0; p < NP; ++p) {
            const float rx = pp[2 * p + 0];
            const float ry = pp[2 * p + 1];
            px[p]  = (rx >= 0.0f) ? rx * 0.5f : -1.0f;   // scale 1024 -> 512
            py[p]  = (ry >= 0.0f) ? ry * 0.5f : -1.0f;
            lab[p] = ll[p];
        }

        // bbox rectangle from first two points unless label[0]==1 (full image).
        float tlx, tly, brx, bry;
        if (lab[0] != 1) {
            tlx = fminf(px[0], px[1]); tly = fminf(py[0], py[1]);
            brx = fmaxf(px[0], px[1]); bry = fmaxf(py[0], py[1]);
        } else {
            tlx = 0.0f; tly = 0.0f; brx = (float)ENC_S; bry = (float)ENC_S;
        }
        const bool rowin = (yf >= tly) && (yf <= bry);   // block-uniform

        // hoist dy^2 per point (block-uniform)
        float dy2[NP];
        #pragma unroll
        for (int p = 0; p < NP; ++p) {
            const float dy = yf - py[p];
            dy2[p] = dy * dy;
        }

        v4f bbox = {0.0f, 0.0f, 0.0f, 0.0f};
        v4f pmsk = {0.0f, 0.0f, 0.0f, 0.0f};
        #pragma unroll
        for (int j = 0; j < 4; ++j) {
            const float xf = xf0 + (float)j;
            const bool colin = (xf >= tlx) && (xf <= brx);
            bbox[j] = (rowin && colin) ? 1.0f : 0.0f;

            bool hit = false;
            #pragma unroll
            for (int p = 0; p < NP; ++p) {
                if (lab[p] == 1) {
                    const float dx = xf - px[p];
                    hit = hit || (dx * dx + dy2[p] <= 25.0f);  // RADIUS^2
                }
            }
            pmsk[j] = hit ? 1.0f : 0.0f;
        }

        float* obase = out + (((size_t)bq * 5) * ENC_S + (size_t)y) * ENC_S + x0;
        #pragma unroll
        for (int c = 0; c < 3; ++c)
            __builtin_nontemporal_store(ch[c], (v4f*)(obase + (size_t)c * ENC_S * ENC_S));
        __builtin_nontemporal_store(bbox, (v4f*)(obase + (size_t)3 * ENC_S * ENC_S));
        __builtin_nontemporal_store(pmsk, (v4f*)(obase + (size_t)4 * ENC_S * ENC_S));
    }
}

extern "C" void kernel_launch(void* const* d_in, const int* in_sizes, int n_in,
                              void* d_out, int out_size, void* d_ws, size_t ws_size,
                              hipStream_t stream) {
    const float* img    = (const float*)d_in[0];   // [B,3,1024,1024] fp32
    const float* pts    = (const float*)d_in[1];   // [B,8,4,2]       fp32
    const int*   labels = (const int*)  d_in[2];   // [B,8,4]         int32
    const float* mean   = (const float*)d_in[3];   // [1,3,1,1]       fp32
    const float* stdv   = (const float*)d_in[4];   // [1,3,1,1]       fp32
    float*       out    = (float*)d_out;           // [B*8,5,512,512] fp32

    const int B = in_sizes[0] / (3 * IN_DIM * IN_DIM);

    dim3 grid(ENC_S, (unsigned)B);   // one block per (output row, batch)
    dim3 block(128);                 // 4 wave32 waves; 4 output pixels/thread
    early_fusion_kernel<<<grid, block, 0, stream>>>(img, pts, labels, mean, stdv, out);
}